// Dep_Context_80109730005366
// MI455X (gfx1250) — compile-verified
//
#include <hip/hip_runtime.h>
#include <hip/hip_bf16.h>

// ---------------------------------------------------------------------------
// Dep_Context on MI455X (gfx1250, wave32).
// Algebraic refactor: energy = q·key has no softmax, so
//   ctx  = q · (key·value)            (KV: [n,10,256], computed once)
//   proj = Mh · (Wp·ctx) · Mw^T       (1x1 conv commutes with bilinear interp)
//   Wp·ctx = q · (KV·Wp^T) = q · A    (A: [n,10,10] per part)
// Total FLOPs ~3e8, traffic ~55 MB -> pure HBM-bandwidth problem (~2.5us at
// 23.3 TB/s). GEMMs use V_WMMA_F32_16X16X4_F32 (full f32 precision), with all
// inner loops branch-free (zero-padded operands, masks instead of exec splits).
// ---------------------------------------------------------------------------

typedef float v2f __attribute__((ext_vector_type(2)));
typedef float v4f __attribute__((ext_vector_type(4)));
typedef float v8f __attribute__((ext_vector_type(8)));

#define N_    4
#define C_    256
#define H_    96
#define W_    96
#define HP    48
#define WP    48
#define HW    (HP * WP)   // 2304
#define HID   10
#define PARTS 6
#define KEYIN 266         // C_ + HID
#define KPAD  272         // key_W K padded for LDS staging
#define EPS_  1e-5f

// ---- Pass 1: 2x2 maxpool of p_fea -> pf [n, c, 48, 48] (2 outputs/thread) --
__global__ void k_maxpool_pf(const float* __restrict__ p_fea, float* __restrict__ pf) {
    int idx = blockIdx.x * blockDim.x + threadIdx.x;
    const int total = N_ * C_ * HW / 2;
    if (idx >= total) return;
    int xp = idx % (WP / 2);
    int t  = idx / (WP / 2);
    int y  = t % HP;
    int nc = t / HP;
    const float* s0 = p_fea + ((size_t)nc * H_ + 2 * y) * W_ + 4 * xp;  // 16B aligned
    v4f r0 = *(const v4f*)s0;
    v4f r1 = *(const v4f*)(s0 + W_);
    v2f o;
    o.x = fmaxf(fmaxf(r0.x, r0.y), fmaxf(r1.x, r1.y));
    o.y = fmaxf(fmaxf(r0.z, r0.w), fmaxf(r1.z, r1.w));
    *(v2f*)(pf + (size_t)nc * HW + y * WP + 2 * xp) = o;
}

// ---- Pass 2: coord features projected by coord_W -> cf [10, 48, 48] --------
__global__ void k_coord(const float* __restrict__ cw, const float* __restrict__ cb,
                        float* __restrict__ cf) {
    int idx = blockIdx.x * blockDim.x + threadIdx.x;
    const int total = HID * HW;
    if (idx >= total) return;
    int x = idx % WP;
    int y = (idx / WP) % HP;
    int o = idx / HW;
    float xmin = x * (2.0f / WP) - 1.0f;
    float xmax = (x + 1) * (2.0f / WP) - 1.0f;
    float ymin = y * (2.0f / HP) - 1.0f;
    float ymax = (y + 1) * (2.0f / HP) - 1.0f;
    float xctr = 0.5f * (xmin + xmax);
    float yctr = 0.5f * (ymin + ymax);
    const float* w = cw + o * 8;
    float s = cb[o];
    s = fmaf(w[0], xmin, s);
    s = fmaf(w[1], ymin, s);
    s = fmaf(w[2], xmax, s);
    s = fmaf(w[3], ymax, s);
    s = fmaf(w[4], xctr, s);
    s = fmaf(w[5], yctr, s);
    s = fmaf(w[6], 1.0f / WP, s);
    s = fmaf(w[7], 1.0f / HP, s);
    cf[idx] = s;
}

// ---- Pass 3: key[n,16(pad),2304] = key_W[10x266] @ [pf;cf] + key_b ---------
// 8 waves/block share an LDS copy of key_W zero-padded to [16][KPAD]; each
// wave computes one 16(o) x 16(x) WMMA tile. Inner loop is branch-free:
// ds_load_b64 (A) + 2x coalesced global_load_b32 (B) + v_wmma per K-step.
__global__ void __launch_bounds__(256) k_key_wmma(
        const float* __restrict__ pf, const float* __restrict__ cf,
        const float* __restrict__ keyW, const float* __restrict__ keyb,
        float* __restrict__ key) {
    __shared__ float sW[16 * KPAD];
    int tid = threadIdx.x;
    for (int i = tid; i < 16 * KPAD; i += 256) {
        int row = i / KPAD, col = i % KPAD;
        sW[i] = (row < HID && col < KEYIN) ? keyW[row * KEYIN + col] : 0.0f;
    }
    __syncthreads();

    int wave = tid >> 5;
    int lane = tid & 31;
    int half = lane >> 4;
    int l16  = lane & 15;
    int n    = blockIdx.y;
    int xcol = (blockIdx.x * 8 + wave) * 16 + l16;
    const float* pfn = pf + (size_t)n * C_ * HW + xcol;

    // cf tail (K = 256..267), zero padded, compile-time register indices
    float ct[12];
#pragma unroll
    for (int j = 0; j < 12; ++j) ct[j] = (j < HID) ? cf[j * HW + xcol] : 0.0f;

    const float* arow = sW + l16 * KPAD + 2 * half;
    v8f acc = {};
    for (int k0 = 0; k0 < C_; k0 += 4) {                    // 64 guard-free steps
        v2f a = *(const v2f*)(arow + k0);                   // 8B-aligned LDS
        v2f b;
        b.x = pfn[(size_t)(k0 + 2 * half) * HW];
        b.y = pfn[(size_t)(k0 + 2 * half + 1) * HW];
        acc = __builtin_amdgcn_wmma_f32_16x16x4_f32(false, a, false, b, (short)0, acc,
                                                    false, false);
    }
#pragma unroll
    for (int k0 = 0; k0 < 12; k0 += 4) {                    // cf tail, unconditional
        v2f a = *(const v2f*)(arow + C_ + k0);
        v2f b;
        b.x = half ? ct[k0 + 2] : ct[k0 + 0];
        b.y = half ? ct[k0 + 3] : ct[k0 + 1];
        acc = __builtin_amdgcn_wmma_f32_16x16x4_f32(false, a, false, b, (short)0, acc,
                                                    false, false);
    }
    // store padded to 16 rows; rows >= 10 written as exact zeros
    float* keyn = key + (size_t)n * 16 * HW + xcol;
#pragma unroll
    for (int r = 0; r < 8; ++r) {
        int orow = r + 8 * half;
        keyn[(size_t)orow * HW] = (orow < HID) ? (acc[r] + keyb[orow]) : 0.0f;
    }
}

// ---- Pass 4: KV[n,16,256] = key[n,16,2304] @ pf^T  (branch-free WMMA) ------
__global__ void k_kv_wmma(const float* __restrict__ pf, const float* __restrict__ key,
                          float* __restrict__ kv) {
    int lane = threadIdx.x;
    int half = lane >> 4;
    int l16  = lane & 15;
    int n  = blockIdx.y;
    int ch = blockIdx.x * 16 + l16;
    const float* arow = key + ((size_t)n * 16 + l16) * HW + 2 * half;  // padded rows
    const float* brow = pf + ((size_t)n * C_ + ch) * HW + 2 * half;
    v8f acc = {};
    for (int x0 = 0; x0 < HW; x0 += 4) {       // two b64 loads + wmma, no branches
        v2f a = *(const v2f*)(arow + x0);
        v2f b = *(const v2f*)(brow + x0);
        acc = __builtin_amdgcn_wmma_f32_16x16x4_f32(false, a, false, b, (short)0, acc,
                                                    false, false);
    }
    float* kvn = kv + (size_t)n * 16 * C_ + ch;
#pragma unroll
    for (int r = 0; r < 8; ++r)
        kvn[(size_t)(r + 8 * half) * C_] = acc[r];         // rows >= 10 are exact 0
}

// ---- Pass 5: A[p,n,16,16] = KV[n] @ proj_W[p]^T  (masked, branch-free) -----
__global__ void k_amat_wmma(const float* __restrict__ kv, const float* __restrict__ projW,
                            float* __restrict__ amat) {
    int lane = threadIdx.x;
    int half = lane >> 4;
    int l16  = lane & 15;
    int p = blockIdx.x, n = blockIdx.y;
    const float* arow = kv + ((size_t)n * 16 + l16) * C_ + 2 * half;
    int   brix  = (l16 < HID) ? l16 : 0;                   // clamp + multiplicative mask
    float bmask = (l16 < HID) ? 1.0f : 0.0f;
    const float* brow = projW + ((size_t)p * HID + brix) * C_ + 2 * half;
    v8f acc = {};
    for (int c0 = 0; c0 < C_; c0 += 4) {
        v2f a = *(const v2f*)(arow + c0);
        v2f b = *(const v2f*)(brow + c0);
        b.x *= bmask;
        b.y *= bmask;
        acc = __builtin_amdgcn_wmma_f32_16x16x4_f32(false, a, false, b, (short)0, acc,
                                                    false, false);
    }
#pragma unroll
    for (int r = 0; r < 8; ++r)
        amat[(((size_t)p * N_ + n) * 16 + (r + 8 * half)) * 16 + l16] = acc[r];
}

// ---- Pass 6: fused hu-maxpool + query projection -> q[p,n,10,2304] ---------
__global__ void k_query(const float* __restrict__ hu, const float* __restrict__ cf,
                        const float* __restrict__ qW, const float* __restrict__ qb,
                        float* __restrict__ q) {
    int x = blockIdx.x * blockDim.x + threadIdx.x;
    if (x >= HW) return;
    int n = blockIdx.y, p = blockIdx.z;
    int yy = x / WP, xx = x % WP;
    const float* hub = hu + (((size_t)p * N_ + n) * HID) * (H_ * W_)
                          + (2 * yy) * W_ + 2 * xx;
    float hup[HID], cfv[HID];
#pragma unroll
    for (int j = 0; j < HID; ++j) {
        const float* s = hub + (size_t)j * (H_ * W_);
        v2f t0 = *(const v2f*)s;                     // 8B aligned (2*xx even)
        v2f t1 = *(const v2f*)(s + W_);
        hup[j] = fmaxf(fmaxf(t0.x, t0.y), fmaxf(t1.x, t1.y));
        cfv[j] = cf[j * HW + x];
    }
    float* qo = q + (((size_t)p * N_ + n) * HID) * HW + x;
#pragma unroll
    for (int k = 0; k < HID; ++k) {
        float s = qb[k];
#pragma unroll
        for (int j = 0; j < HID; ++j) {
            s = fmaf(qW[k * 20 + j], hup[j], s);
            s = fmaf(qW[k * 20 + HID + j], cfv[j], s);
        }
        qo[(size_t)k * HW] = s;
    }
}

// ---- Pass 7: small[p,n,o,x] = sum_k A[p,n,k,o] * q[p,n,k,x] ----------------
__global__ void k_small(const float* __restrict__ q, const float* __restrict__ amat,
                        float* __restrict__ small) {
    __shared__ float As[16][16];
    int tid = threadIdx.x;
    int p = blockIdx.z, n = blockIdx.y;
    As[tid >> 4][tid & 15] = amat[(((size_t)p * N_ + n) * 16) * 16 + tid];
    __syncthreads();
    int x = blockIdx.x * blockDim.x + tid;
    if (x >= HW) return;
    const float* qx = q + (((size_t)p * N_ + n) * HID) * HW + x;
    float qv[HID];
#pragma unroll
    for (int k = 0; k < HID; ++k) qv[k] = qx[(size_t)k * HW];
    float* so = small + (((size_t)p * N_ + n) * HID) * HW + x;
#pragma unroll
    for (int o = 0; o < HID; ++o) {
        float s = 0.0f;
#pragma unroll
        for (int k = 0; k < HID; ++k) s = fmaf(As[k][o], qv[k], s);
        so[(size_t)o * HW] = s;
    }
}

// ---- Pass 8: bilinear upsample (align_corners) + BN + ReLU -> out ----------
__global__ void k_up_bn(const float* __restrict__ small, const float* __restrict__ g,
                        const float* __restrict__ beta, const float* __restrict__ mean,
                        const float* __restrict__ var, float* __restrict__ out) {
    int idx = blockIdx.x * blockDim.x + threadIdx.x;
    const int total = PARTS * N_ * HID * H_ * W_;
    if (idx >= total) return;
    int Wc = idx % W_;
    int t  = idx / W_;
    int Hc = t % H_;  t /= H_;
    int o  = t % HID; t /= HID;
    int n  = t % N_;
    int p  = t / N_;
    const float rr = (float)(HP - 1) / (float)(H_ - 1);  // 47/95
    float py = Hc * rr, px = Wc * rr;
    int y0 = (int)py; if (y0 > HP - 1) y0 = HP - 1;
    int x0 = (int)px; if (x0 > WP - 1) x0 = WP - 1;
    float wy = py - y0, wx = px - x0;
    int y1 = y0 + 1; if (y1 > HP - 1) y1 = HP - 1;
    int x1 = x0 + 1; if (x1 > WP - 1) x1 = WP - 1;
    const float* S = small + (((size_t)p * N_ + n) * HID + o) * HW;
    float v00 = S[y0 * WP + x0], v01 = S[y0 * WP + x1];
    float v10 = S[y1 * WP + x0], v11 = S[y1 * WP + x1];
    float val = (1.0f - wy) * fmaf(wx, v01 - v00, v00) + wy * fmaf(wx, v11 - v10, v10);
    int po = p * HID + o;
    float r2 = fmaf(g[po] * rsqrtf(var[po] + EPS_), val - mean[po], beta[po]);
    out[idx] = r2 > 0.0f ? r2 : 0.0f;
}

// ---------------------------------------------------------------------------
extern "C" void kernel_launch(void* const* d_in, const int* in_sizes, int n_in,
                              void* d_out, int out_size, void* d_ws, size_t ws_size,
                              hipStream_t stream) {
    const float* p_fea    = (const float*)d_in[0];
    const float* hu       = (const float*)d_in[1];
    const float* coord_W  = (const float*)d_in[2];
    const float* coord_b  = (const float*)d_in[3];
    const float* query_W  = (const float*)d_in[4];
    const float* query_b  = (const float*)d_in[5];
    const float* key_W    = (const float*)d_in[6];
    const float* key_b    = (const float*)d_in[7];
    const float* proj_W   = (const float*)d_in[8];
    const float* bn_gamma = (const float*)d_in[9];
    const float* bn_beta  = (const float*)d_in[10];
    const float* bn_mean  = (const float*)d_in[11];
    const float* bn_var   = (const float*)d_in[12];
    float* out = (float*)d_out;

    // workspace layout (floats); total ~3.66M floats (~14.6 MB)
    float* ws    = (float*)d_ws;
    float* pf    = ws;                  size_t off = (size_t)N_ * C_ * HW;
    float* cf    = ws + off;            off += (size_t)HID * HW;
    float* key   = ws + off;            off += (size_t)N_ * 16 * HW;     // padded rows
    float* kv    = ws + off;            off += (size_t)N_ * 16 * C_;
    float* q     = ws + off;            off += (size_t)PARTS * N_ * HID * HW;
    float* amat  = ws + off;            off += (size_t)PARTS * N_ * 16 * 16;
    float* small = ws + off;

    {   // Pass 1: maxpool p_fea (2 outputs per thread, b128 loads)
        int total = N_ * C_ * HW / 2;
        k_maxpool_pf<<<(total + 255) / 256, 256, 0, stream>>>(p_fea, pf);
    }
    {   // Pass 2: coord features
        int total = HID * HW;
        k_coord<<<(total + 255) / 256, 256, 0, stream>>>(coord_W, coord_b, cf);
    }
    {   // Pass 3: key (WMMA f32, LDS-staged weights, 8 waves/block)
        dim3 grid(HW / 16 / 8, N_);
        k_key_wmma<<<grid, 256, 0, stream>>>(pf, cf, key_W, key_b, key);
    }
    {   // Pass 4: KV = key @ value (WMMA f32, branch-free)
        dim3 grid(C_ / 16, N_);
        k_kv_wmma<<<grid, 32, 0, stream>>>(pf, key, kv);
    }
    {   // Pass 5: A = KV @ Wp^T (WMMA f32, masked)
        dim3 grid(PARTS, N_);
        k_amat_wmma<<<grid, 32, 0, stream>>>(kv, proj_W, amat);
    }
    {   // Pass 6: fused hu maxpool + query
        dim3 grid((HW + 255) / 256, N_, PARTS);
        k_query<<<grid, 256, 0, stream>>>(hu, cf, query_W, query_b, q);
    }
    {   // Pass 7: small = A . q
        dim3 grid((HW + 255) / 256, N_, PARTS);
        k_small<<<grid, 256, 0, stream>>>(q, amat, small);
    }
    {   // Pass 8: upsample + BN + ReLU
        int total = PARTS * N_ * HID * H_ * W_;
        k_up_bn<<<(total + 255) / 256, 256, 0, stream>>>(small, bn_gamma, bn_beta,
                                                         bn_mean, bn_var, out);
    }
}